// LossBuilder_76553497084344
// MI455X (gfx1250) — compile-verified
//
#include <hip/hip_runtime.h>

#define N_ROWS 4096
#define D_DIM  2048
#define C_CLS  8192
#define MARGIN_F 0.3f
#define EPS_LS   0.1f

#define LDSS 36   // padded row stride (floats): 144 B, 16B-aligned, conflict-free

typedef float v2f __attribute__((ext_vector_type(2)));
typedef float v8f __attribute__((ext_vector_type(8)));

// ---------------------------------------------------------------------------
// Init: d_out[0..1] = 0, dist_ap = -inf, dist_an = +inf
// ---------------------------------------------------------------------------
__global__ void init_out_kernel(float* __restrict__ out) {
    int i = blockIdx.x * blockDim.x + threadIdx.x;
    if (i >= 2 + 2 * N_ROWS) return;
    if (i < 2)               out[i] = 0.0f;
    else if (i < 2 + N_ROWS) out[i] = -__builtin_inff();   // dist_ap
    else                     out[i] =  __builtin_inff();   // dist_an
}

// ---------------------------------------------------------------------------
// Row squared-norms of embeddings: norms[i] = sum_k E[i,k]^2
// ---------------------------------------------------------------------------
__global__ __launch_bounds__(256)
void row_norms_kernel(const float* __restrict__ E, float* __restrict__ norms) {
    __shared__ float red[256];
    const int row = blockIdx.x;
    const int tid = threadIdx.x;
    const float* p = E + (size_t)row * D_DIM + tid * 8;
    float4 a = *(const float4*)(p);
    float4 b = *(const float4*)(p + 4);
    float s = a.x*a.x + a.y*a.y + a.z*a.z + a.w*a.w
            + b.x*b.x + b.y*b.y + b.z*b.z + b.w*b.w;
    red[tid] = s;
    __syncthreads();
    #pragma unroll
    for (int st = 128; st > 0; st >>= 1) {
        if (tid < st) red[tid] += red[tid + st];
        __syncthreads();
    }
    if (tid == 0) norms[row] = red[0];
}

// ---------------------------------------------------------------------------
// Fused fp32 Gram (v_wmma_f32_16x16x4_f32) + euclidean dist + batch-hard
// mining.  Workgroup tile 128x128, 8 waves, each wave owns a 32x64 strip
// (2 M-frags x 4 N-frags => 8 accumulators).  K staged in 32-wide LDS tiles,
// DOUBLE-BUFFERED via global_load_async_to_lds_b128 (ASYNCcnt) so global
// latency hides behind the WMMA stream.
// ---------------------------------------------------------------------------
__global__ __launch_bounds__(256)
void gram_mine_kernel(const float* __restrict__ E,
                      const float* __restrict__ norms,
                      const int*   __restrict__ tgt,
                      float* __restrict__ dist_ap,
                      float* __restrict__ dist_an) {
    __shared__ float As[2][128 * LDSS];
    __shared__ float Bs[2][128 * LDSS];
    __shared__ float apS[128];
    __shared__ float anS[128];

    const int tid  = threadIdx.x;
    const int lane = tid & 31;
    const int wave = tid >> 5;       // 0..7
    const int wm   = wave >> 1;      // 0..3 : rows wm*32 .. +32
    const int wn   = wave & 1;       // 0..1 : cols wn*64 .. +64
    const int hi   = lane >> 4;
    const int l15  = lane & 15;

    const unsigned rowBase = blockIdx.y * 128;
    const unsigned colBase = blockIdx.x * 128;

    if (tid < 128) { apS[tid] = -__builtin_inff(); anS[tid] = __builtin_inff(); }

    // per-thread staging geometry: 4 b128 rows for A, 4 for B per K-tile
    const unsigned r_   = (unsigned)(tid >> 3);          // 0..31 (+32*i)
    const unsigned c4_  = (unsigned)((tid & 7) << 2);    // 0,4,..,28

    // issue one K-tile's async global->LDS copies (8 instrs per thread/wave)
    auto issue_tile = [&](int buf, unsigned k0) {
        #pragma unroll
        for (int i = 0; i < 4; ++i) {
            unsigned r  = r_ + i * 32;
            unsigned ga = ((rowBase + r) * D_DIM + k0 + c4_) * 4u;
            unsigned la = (unsigned)(uintptr_t)(&As[buf][r * LDSS + c4_]);
            asm volatile("global_load_async_to_lds_b128 %0, %1, %2"
                         :: "v"(la), "v"(ga), "s"(E) : "memory");
            unsigned gb = ((colBase + r) * D_DIM + k0 + c4_) * 4u;
            unsigned lb = (unsigned)(uintptr_t)(&Bs[buf][r * LDSS + c4_]);
            asm volatile("global_load_async_to_lds_b128 %0, %1, %2"
                         :: "v"(lb), "v"(gb), "s"(E) : "memory");
        }
    };

    v8f acc[2][4] = {};              // [m-frag][n-frag], 16x16 fp32 each

    issue_tile(0, 0);

    for (unsigned k0 = 0; k0 < D_DIM; k0 += 32) {
        const int cur = (k0 >> 5) & 1;
        if (k0 + 32 < D_DIM) {
            issue_tile(cur ^ 1, k0 + 32);
            // current tile's 8 ops retire first (async loads are in-order):
            asm volatile("s_wait_asynccnt 8" ::: "memory");
        } else {
            asm volatile("s_wait_asynccnt 0" ::: "memory");
        }
        __syncthreads();

        const float* Ac = &As[cur][0];
        const float* Bc = &Bs[cur][0];
        #pragma unroll
        for (int kk = 0; kk < 32; kk += 4) {
            v2f a0 = *(const v2f*)(Ac + (wm * 32 +      l15) * LDSS + kk + 2 * hi);
            v2f a1 = *(const v2f*)(Ac + (wm * 32 + 16 + l15) * LDSS + kk + 2 * hi);
            #pragma unroll
            for (int f = 0; f < 4; ++f) {
                v2f b = *(const v2f*)(Bc + (wn * 64 + f * 16 + l15) * LDSS + kk + 2 * hi);
                acc[0][f] = __builtin_amdgcn_wmma_f32_16x16x4_f32(
                    false, a0, false, b, (short)0, acc[0][f], false, false);
                acc[1][f] = __builtin_amdgcn_wmma_f32_16x16x4_f32(
                    false, a1, false, b, (short)0, acc[1][f], false, false);
            }
        }
        __syncthreads();
    }

    // ---- epilogue: dist = sqrt(max(xx + yy - 2*dot, 1e-12)), mine ----
    float cxx[4];
    int   ctg[4];
    #pragma unroll
    for (int f = 0; f < 4; ++f) {
        int gc = colBase + wn * 64 + f * 16 + l15;   // C/D layout: N = l15
        cxx[f] = norms[gc];
        ctg[f] = tgt[gc];
    }
    #pragma unroll
    for (int mf = 0; mf < 2; ++mf) {
        #pragma unroll
        for (int v = 0; v < 8; ++v) {
            int lr  = wm * 32 + mf * 16 + v + 8 * hi; // C/D layout: M = v + 8*hi
            int gr  = rowBase + lr;
            float rxx = norms[gr];
            int   rtg = tgt[gr];
            float pmax = -__builtin_inff();
            float nmin =  __builtin_inff();
            #pragma unroll
            for (int f = 0; f < 4; ++f) {
                float d2   = rxx + cxx[f] - 2.0f * acc[mf][f][v];
                float dist = sqrtf(fmaxf(d2, 1e-12f));
                if (rtg == ctg[f]) pmax = fmaxf(pmax, dist);
                else               nmin = fminf(nmin, dist);
            }
            if (pmax > -__builtin_inff())
                __hip_atomic_fetch_max(&apS[lr], pmax, __ATOMIC_RELAXED,
                                       __HIP_MEMORY_SCOPE_WORKGROUP);
            if (nmin < __builtin_inff())
                __hip_atomic_fetch_min(&anS[lr], nmin, __ATOMIC_RELAXED,
                                       __HIP_MEMORY_SCOPE_WORKGROUP);
        }
    }
    __syncthreads();
    if (tid < 128) {
        int gr = rowBase + tid;
        __hip_atomic_fetch_max(&dist_ap[gr], apS[tid], __ATOMIC_RELAXED,
                               __HIP_MEMORY_SCOPE_AGENT);
        __hip_atomic_fetch_min(&dist_an[gr], anS[tid], __ATOMIC_RELAXED,
                               __HIP_MEMORY_SCOPE_AGENT);
    }
}

// ---------------------------------------------------------------------------
// Label-smoothed cross entropy, one workgroup per row; registers hold the
// row so HBM is read exactly once (memory-bound: 128 MB @ 23.3 TB/s).
// ---------------------------------------------------------------------------
__global__ __launch_bounds__(256)
void ce_kernel(const float* __restrict__ P, const int* __restrict__ tgt,
               float* __restrict__ out0) {
    __shared__ float red[256];
    __shared__ float xt_s;
    const int row = blockIdx.x;
    const int tid = threadIdx.x;
    const float* p = P + (size_t)row * C_CLS + tid * 32;

    float x[32];
    #pragma unroll
    for (int j = 0; j < 8; ++j) {
        float4 v = *(const float4*)(p + j * 4);
        x[j*4+0] = v.x; x[j*4+1] = v.y; x[j*4+2] = v.z; x[j*4+3] = v.w;
    }

    const int t = tgt[row];
    if (tid == (t >> 5)) xt_s = x[t & 31];

    float m = -__builtin_inff();
    #pragma unroll
    for (int j = 0; j < 32; ++j) m = fmaxf(m, x[j]);
    red[tid] = m;
    __syncthreads();
    #pragma unroll
    for (int st = 128; st > 0; st >>= 1) {
        if (tid < st) red[tid] = fmaxf(red[tid], red[tid + st]);
        __syncthreads();
    }
    m = red[0];
    __syncthreads();

    float se = 0.0f, sx = 0.0f;
    #pragma unroll
    for (int j = 0; j < 32; ++j) { se += expf(x[j] - m); sx += x[j]; }
    red[tid] = se;
    __syncthreads();
    #pragma unroll
    for (int st = 128; st > 0; st >>= 1) {
        if (tid < st) red[tid] += red[tid + st];
        __syncthreads();
    }
    float se_tot = red[0];
    __syncthreads();

    red[tid] = sx;
    __syncthreads();
    #pragma unroll
    for (int st = 128; st > 0; st >>= 1) {
        if (tid < st) red[tid] += red[tid + st];
        __syncthreads();
    }
    float sx_tot = red[0];

    if (tid == 0) {
        float lse    = m + logf(se_tot);
        float nll    = lse - xt_s;
        float smooth = lse - sx_tot / (float)C_CLS;
        float lrow   = (1.0f - EPS_LS) * nll + EPS_LS * smooth;
        __hip_atomic_fetch_add(out0, lrow / (float)N_ROWS, __ATOMIC_RELAXED,
                               __HIP_MEMORY_SCOPE_AGENT);
    }
}

// ---------------------------------------------------------------------------
// tri_loss = mean(relu(dist_ap - dist_an + margin)); single workgroup
// ---------------------------------------------------------------------------
__global__ __launch_bounds__(256)
void tri_kernel(const float* __restrict__ ap, const float* __restrict__ an,
                float* __restrict__ out1) {
    __shared__ float red[256];
    const int tid = threadIdx.x;
    float s = 0.0f;
    for (int i = tid; i < N_ROWS; i += 256)
        s += fmaxf(ap[i] - an[i] + MARGIN_F, 0.0f);
    red[tid] = s;
    __syncthreads();
    #pragma unroll
    for (int st = 128; st > 0; st >>= 1) {
        if (tid < st) red[tid] += red[tid + st];
        __syncthreads();
    }
    if (tid == 0) out1[0] = red[0] / (float)N_ROWS;
}

// ---------------------------------------------------------------------------
extern "C" void kernel_launch(void* const* d_in, const int* in_sizes, int n_in,
                              void* d_out, int out_size, void* d_ws, size_t ws_size,
                              hipStream_t stream) {
    const float* E = (const float*)d_in[0];   // embeddings [4096, 2048] fp32
    const float* P = (const float*)d_in[1];   // pred_ids   [4096, 8192] fp32
    const int*   T = (const int*)d_in[2];     // target_ids [4096] int32

    float* out     = (float*)d_out;           // [id, tri, ap[4096], an[4096]]
    float* norms   = (float*)d_ws;            // 4096 fp32 scratch
    float* dist_ap = out + 2;
    float* dist_an = out + 2 + N_ROWS;

    init_out_kernel<<<(2 + 2 * N_ROWS + 255) / 256, 256, 0, stream>>>(out);
    row_norms_kernel<<<N_ROWS, 256, 0, stream>>>(E, norms);

    dim3 grid(N_ROWS / 128, N_ROWS / 128);    // (col tiles, row tiles)
    gram_mine_kernel<<<grid, 256, 0, stream>>>(E, norms, T, dist_ap, dist_an);

    ce_kernel<<<N_ROWS, 256, 0, stream>>>(P, T, out);
    tri_kernel<<<1, 256, 0, stream>>>(dist_ap, dist_an, out + 1);
}